// TopKPoolBroadcastGCN_49615462204215
// MI455X (gfx1250) — compile-verified
//
#include <hip/hip_runtime.h>
#include <hip/hip_bf16.h>

typedef __attribute__((ext_vector_type(2))) float v2f;
typedef __attribute__((ext_vector_type(8))) float v8f;

#define FDIM 128

struct SelState {
  unsigned prefix;
  int krem;
  unsigned thr;
  int tiequota;
  int maxdeg;
  int pad0;
  unsigned long long bestglob;
  int fallback;
  int pad1;
};

// ---------------- utility ----------------
__global__ void k_fill0(unsigned* p, size_t nwords) {
  size_t i = (size_t)blockIdx.x * blockDim.x + threadIdx.x;
  if (i < nwords) p[i] = 0u;
}

__global__ void k_init(SelState* st, int K) {
  if (threadIdx.x == 0 && blockIdx.x == 0) st->krem = K;
}

// ---------------- fp32 WMMA GEMM: C[M x 128] = A[M x 128] @ W[128 x 128] ----------------
// One 256-thread block = 8 waves; wave w computes the 16x16 tile (blockIdx.x, w).
// Fully unrolled K: 32x V_WMMA_F32_16X16X4_F32 per wave, branch-free inner loop
// (partial tiles handled by clamping the A row: C row m depends only on A row m,
//  and out-of-range C rows are simply not stored).
__global__ void k_gemm128(const float* __restrict__ A, const float* __restrict__ W,
                          float* __restrict__ C, int M) {
  int wave = threadIdx.x >> 5;
  int lane = threadIdx.x & 31;
  int lr = lane & 15;      // 0..15
  int hi = lane >> 4;      // 0 or 1 (selects K pair per ISA A-frag layout)
  int m0 = blockIdx.x * 16;
  int n0 = wave * 16;
  int rowA = m0 + lr;
  if (rowA >= M) rowA = M - 1;                 // branchless clamp (v_min)
  const float* Arow = A + (size_t)rowA * FDIM + hi * 2;   // 8B aligned (even K)
  const float* Wcol = W + (size_t)(hi * 2) * FDIM + n0 + lr;
  v8f acc = {};
#pragma unroll
  for (int k0 = 0; k0 < FDIM; k0 += 4) {
    v2f a = *(const v2f*)(Arow + k0);          // global_load_b64
    v2f b;
    b.x = Wcol[(size_t)k0 * FDIM];
    b.y = Wcol[(size_t)k0 * FDIM + FDIM];
    acc = __builtin_amdgcn_wmma_f32_16x16x4_f32(false, a, false, b,
                                                (short)0, acc, false, false);
  }
  // C layout: VGPR r -> M = m0 + hi*8 + r, N = n0 + lr
  float* Cbase = C + n0 + lr;
  if (m0 + 16 <= M) {
#pragma unroll
    for (int r = 0; r < 8; ++r)
      Cbase[(size_t)(m0 + hi * 8 + r) * FDIM] = acc[r];
  } else {
#pragma unroll
    for (int r = 0; r < 8; ++r) {
      int rowC = m0 + hi * 8 + r;
      if (rowC < M) Cbase[(size_t)rowC * FDIM] = acc[r];
    }
  }
}

// ---------------- conv1 degree / normalization ----------------
__global__ void k_deg_col(const int* __restrict__ col, float* __restrict__ degf, int E) {
  int e = blockIdx.x * blockDim.x + threadIdx.x;
  if (e < E) atomicAdd(&degf[col[e]], 1.0f);
}

__global__ void k_rsqrt1p(const float* __restrict__ d, float* __restrict__ dinv, int n) {
  int i = blockIdx.x * blockDim.x + threadIdx.x;
  if (i < n) dinv[i] = rsqrtf(d[i] + 1.0f);
}

// agg[col] += dinv[row]*dinv[col]*H[row]  (one wave per edge, 128 feats)
__global__ void k_agg1(const int* __restrict__ row, const int* __restrict__ col,
                       const float* __restrict__ dinv, const float* __restrict__ H,
                       float* __restrict__ agg, int E) {
  int t = blockIdx.x * blockDim.x + threadIdx.x;
  int e = t >> 5, lane = t & 31;
  if (e >= E) return;
  int r = row[e], c = col[e];
  float nrm = dinv[r] * dinv[c];
  const float* hs = H + (size_t)r * FDIM;
  float* ad = agg + (size_t)c * FDIM;
  for (int f = lane; f < FDIM; f += 32) atomicAdd(&ad[f], nrm * hs[f]);
}

// x1 = relu(agg + dinv^2 * H + b1), in place into agg
__global__ void k_self_relu(float* __restrict__ x1io, const float* __restrict__ H,
                            const float* __restrict__ dinv, const float* __restrict__ b1,
                            int N) {
  int idx = blockIdx.x * blockDim.x + threadIdx.x;
  if (idx >= N * FDIM) return;
  int i = idx >> 7, f = idx & (FDIM - 1);
  float v = x1io[idx] + dinv[i] * dinv[i] * H[idx] + b1[f];
  x1io[idx] = v > 0.0f ? v : 0.0f;
}

// raw = x1 . w_score ; x1g = x1*tanh(raw) ; monotone uint key of raw
__global__ void k_score_gate(const float* __restrict__ x1, const float* __restrict__ wsc,
                             float* __restrict__ x1g, unsigned* __restrict__ ukey, int N) {
  int node = blockIdx.x * (blockDim.x >> 5) + (threadIdx.x >> 5);
  int lane = threadIdx.x & 31;
  if (node >= N) return;
  const float* xr = x1 + (size_t)node * FDIM;
  float s = 0.0f;
  for (int f = lane; f < FDIM; f += 32) s += xr[f] * wsc[f];
  for (int off = 16; off > 0; off >>= 1) s += __shfl_down(s, off);
  s = __shfl(s, 0);
  float t = tanhf(s);
  float* gr = x1g + (size_t)node * FDIM;
  for (int f = lane; f < FDIM; f += 32) gr[f] = xr[f] * t;
  if (lane == 0) {
    unsigned u = __float_as_uint(s);
    u = (u & 0x80000000u) ? ~u : (u | 0x80000000u);
    ukey[node] = u;
  }
}

// ---------------- top-K radix select (4 byte passes, descending) ----------------
__global__ void k_hist(const unsigned* __restrict__ u, int* __restrict__ hist,
                       const SelState* __restrict__ st, int pass, int N) {
  int i = blockIdx.x * blockDim.x + threadIdx.x;
  if (i >= N) return;
  unsigned v = u[i];
  if (pass > 0 && (v >> (32 - 8 * pass)) != st->prefix) return;
  atomicAdd(&hist[(v >> (24 - 8 * pass)) & 255], 1);
}

__global__ void k_pick(SelState* st, int* hist, int pass) {
  if (threadIdx.x || blockIdx.x) return;
  int krem = st->krem;
  unsigned pref = st->prefix;
  int cum = 0, chosen = 0, greater = 0;
  for (int b = 255; b >= 0; --b) {
    int h = hist[b];
    if (cum + h >= krem) { chosen = b; greater = cum; break; }
    cum += h;
  }
  st->prefix = (pref << 8) | (unsigned)chosen;
  st->krem = krem - greater;
  for (int b = 0; b < 256; ++b) hist[b] = 0;
  if (pass == 3) { st->thr = st->prefix; st->tiequota = st->krem; }
}

// keep mask + cluster labels in index order (single block, deterministic tie quota)
__global__ void k_label(const unsigned* __restrict__ u, const SelState* __restrict__ st,
                        int* __restrict__ keep, int* __restrict__ cof, int N) {
  __shared__ int sh[256];
  __shared__ int running, quota;
  unsigned thr = st->thr;
  if (threadIdx.x == 0) { running = 0; quota = st->tiequota; }
  __syncthreads();
  for (int base = 0; base < N; base += 256) {
    int i = base + threadIdx.x;
    int f = 0;
    if (i < N) {
      unsigned v = u[i];
      f = (v > thr) ? 1 : ((v == thr) ? 2 : 0);
    }
    sh[threadIdx.x] = f;
    __syncthreads();
    if (threadIdx.x == 0) {
      int s = running, q = quota;
      for (int t = 0; t < 256; ++t) {
        int ff = sh[t], kept = 0;
        if (ff == 1) kept = 1;
        else if (ff == 2 && q > 0) { kept = 1; --q; }
        sh[t] = kept ? s : -1;
        s += kept;
      }
      running = s; quota = q;
    }
    __syncthreads();
    if (i < N) { int c = sh[threadIdx.x]; keep[i] = (c >= 0); cof[i] = c; }
    __syncthreads();
  }
}

// ---------------- cluster assignment ----------------
__global__ void k_deg_row(const int* __restrict__ row, int* __restrict__ deg, int E) {
  int e = blockIdx.x * blockDim.x + threadIdx.x;
  if (e < E) atomicAdd(&deg[row[e]], 1);
}

__global__ void k_bestglob1(const int* __restrict__ keep, const int* __restrict__ deg,
                            SelState* st, int N) {
  int i = blockIdx.x * blockDim.x + threadIdx.x;
  if (i < N && keep[i]) atomicMax(&st->maxdeg, deg[i]);
}

__global__ void k_bestglob2(const int* __restrict__ keep, const int* __restrict__ deg,
                            const unsigned* __restrict__ u, SelState* st, int N) {
  int i = blockIdx.x * blockDim.x + threadIdx.x;
  if (i >= N || !keep[i] || deg[i] != st->maxdeg) return;
  unsigned long long key = ((unsigned long long)u[i] << 17) | (unsigned)(131071 - i);
  atomicMax(&st->bestglob, key);
}

// composite key: deg desc, earliest insertion; packed (valid|comp<<17|neigh)
__global__ void k_edge_best(const int* __restrict__ row, const int* __restrict__ col,
                            const int* __restrict__ deg, const int* __restrict__ keep,
                            unsigned long long* __restrict__ best, int E) {
  int e = blockIdx.x * blockDim.x + threadIdx.x;
  if (e >= E) return;
  unsigned long long M = 2ULL * (unsigned long long)E + 2ULL;
  int r = row[e], c = col[e];
  if (keep[c]) {
    unsigned long long comp = (unsigned long long)deg[c] * M + (M - 1 - 2ULL * (unsigned long long)e);
    atomicMax(&best[r], (1ULL << 63) | (comp << 17) | (unsigned long long)(unsigned)c);
  }
  if (keep[r]) {
    unsigned long long comp = (unsigned long long)deg[r] * M + (M - 2 - 2ULL * (unsigned long long)e);
    atomicMax(&best[c], (1ULL << 63) | (comp << 17) | (unsigned long long)(unsigned)r);
  }
}

__global__ void k_fallback(SelState* st, const int* __restrict__ cof) {
  if (threadIdx.x == 0 && blockIdx.x == 0)
    st->fallback = cof[131071 - (int)(st->bestglob & 0x1FFFFull)];
}

__global__ void k_assign(const int* __restrict__ keep, const int* __restrict__ cof,
                         const unsigned long long* __restrict__ best,
                         const SelState* __restrict__ st, int* __restrict__ cid,
                         float* __restrict__ counts, int N) {
  int i = blockIdx.x * blockDim.x + threadIdx.x;
  if (i >= N) return;
  int c;
  if (keep[i]) c = cof[i];
  else {
    unsigned long long b = best[i];
    c = (b >> 63) ? cof[(int)(b & 0x1FFFFull)] : st->fallback;
  }
  cid[i] = c;
  atomicAdd(&counts[c], 1.0f);
}

// ---------------- pooling ----------------
__global__ void k_pool(const int* __restrict__ cid, const float* __restrict__ x1g,
                       float* __restrict__ sums, int N) {
  int t = blockIdx.x * blockDim.x + threadIdx.x;
  int i = t >> 5, lane = t & 31;
  if (i >= N) return;
  int c = cid[i];
  const float* src = x1g + (size_t)i * FDIM;
  float* dst = sums + (size_t)c * FDIM;
  for (int f = lane; f < FDIM; f += 32) atomicAdd(&dst[f], src[f]);
}

__global__ void k_meandiv(const float* __restrict__ sums, const float* __restrict__ counts,
                          float* __restrict__ xp, int K) {
  int idx = blockIdx.x * blockDim.x + threadIdx.x;
  if (idx >= K * FDIM) return;
  int c = idx >> 7;
  xp[idx] = sums[idx] / fmaxf(counts[c], 1.0f);
}

// ---------------- coarse GCN (conv2) ----------------
__global__ void k_deg2(const int* __restrict__ row, const int* __restrict__ col,
                       const int* __restrict__ cid, float* __restrict__ deg2, int E) {
  int e = blockIdx.x * blockDim.x + threadIdx.x;
  if (e >= E) return;
  int cu = cid[row[e]], cv = cid[col[e]];
  if (cu != cv) atomicAdd(&deg2[cv], 1.0f);
}

__global__ void k_agg2(const int* __restrict__ row, const int* __restrict__ col,
                       const int* __restrict__ cid, const float* __restrict__ dinv2,
                       const float* __restrict__ h2, float* __restrict__ agg2, int E) {
  int t = blockIdx.x * blockDim.x + threadIdx.x;
  int e = t >> 5, lane = t & 31;
  if (e >= E) return;
  int cu = cid[row[e]], cv = cid[col[e]];
  if (cu == cv) return;
  float nrm = dinv2[cu] * dinv2[cv];
  const float* hs = h2 + (size_t)cu * FDIM;
  float* ad = agg2 + (size_t)cv * FDIM;
  for (int f = lane; f < FDIM; f += 32) atomicAdd(&ad[f], nrm * hs[f]);
}

__global__ void k_self2(float* __restrict__ xp2io, const float* __restrict__ h2,
                        const float* __restrict__ dinv2, const float* __restrict__ b2,
                        int K) {
  int idx = blockIdx.x * blockDim.x + threadIdx.x;
  if (idx >= K * FDIM) return;
  int c = idx >> 7, f = idx & (FDIM - 1);
  xp2io[idx] = xp2io[idx] + dinv2[c] * dinv2[c] * h2[idx] + b2[f];
}

// ---------------- broadcast + skip + bias, plus scalar 0.0 output ----------------
__global__ void k_final(float* __restrict__ out, const float* __restrict__ xp2,
                        const int* __restrict__ cid, const float* __restrict__ bskip,
                        int N) {
  int idx = blockIdx.x * blockDim.x + threadIdx.x;
  if (idx >= N * FDIM) return;
  int i = idx >> 7, f = idx & (FDIM - 1);
  out[idx] = out[idx] + xp2[(size_t)cid[i] * FDIM + f] + bskip[f];
  if (idx == 0) out[(size_t)N * FDIM] = 0.0f;
}

// ---------------- host-side orchestration ----------------
extern "C" void kernel_launch(void* const* d_in, const int* in_sizes, int n_in,
                              void* d_out, int out_size, void* d_ws, size_t ws_size,
                              hipStream_t stream) {
  const float* x     = (const float*)d_in[0];
  const int*   ei    = (const int*)d_in[1];
  const float* W1    = (const float*)d_in[2];
  const float* b1    = (const float*)d_in[3];
  const float* W2    = (const float*)d_in[4];
  const float* b2    = (const float*)d_in[5];
  const float* wsc   = (const float*)d_in[6];
  const float* Wskip = (const float*)d_in[7];
  const float* bskip = (const float*)d_in[8];
  float* out = (float*)d_out;

  const int N = in_sizes[0] / FDIM;
  const int E = in_sizes[1] / 2;
  const int K = (N < 5000) ? N : 5000;
  const int* rowp = ei;
  const int* colp = ei + E;

  const size_t NF = (size_t)N * FDIM;
  const size_t KF = (size_t)K * FDIM;

  char* w = (char*)d_ws;
  size_t off = 0;
  auto alloc = [&](size_t bytes) -> void* {
    void* p = w + off;
    off = (off + bytes + 255) & ~(size_t)255;
    return p;
  };
  // fully-overwritten buffers (no zeroing needed)
  float*    bufA = (float*)alloc(NF * 4);       // H1, later x1g
  float*    dinv = (float*)alloc((size_t)N * 4);
  unsigned* ukey = (unsigned*)alloc((size_t)N * 4);
  int*      keep = (int*)alloc((size_t)N * 4);
  int*      cof  = (int*)alloc((size_t)N * 4);
  int*      cid  = (int*)alloc((size_t)N * 4);
  float*    x_p  = (float*)alloc(KF * 4);
  float*    h2   = (float*)alloc(KF * 4);
  float*    dinv2= (float*)alloc((size_t)K * 4);
  // atomically-accumulated buffers -> one contiguous zero region
  size_t zstart = off;
  float*    agg  = (float*)alloc(NF * 4);       // agg1, later x1 (in place)
  float*    degf = (float*)alloc((size_t)N * 4);
  int*      degr = (int*)alloc((size_t)N * 4);
  unsigned long long* best = (unsigned long long*)alloc((size_t)N * 8);
  float*    sums = (float*)alloc(KF * 4);
  float*    cnts = (float*)alloc((size_t)K * 4);
  float*    agg2 = (float*)alloc(KF * 4);       // later xp2 (in place)
  float*    deg2 = (float*)alloc((size_t)K * 4);
  int*      hist = (int*)alloc(256 * 4);
  SelState* st   = (SelState*)alloc(sizeof(SelState));
  size_t zwords = (off - zstart) / 4;

  const int B = 256;
  auto g  = [&](long long n) { return (unsigned)((n + B - 1) / B); };

  // 1) zero the accumulation region, init selection state
  k_fill0<<<g((long long)zwords), B, 0, stream>>>((unsigned*)(w + zstart), zwords);
  k_init<<<1, 64, 0, stream>>>(st, K);

  // 2) conv1: H1 = x @ W1 (WMMA f32)
  k_gemm128<<<(N + 15) / 16, B, 0, stream>>>(x, W1, bufA, N);

  // 3) conv1 normalization + aggregation + self-loop + bias + relu
  k_deg_col<<<g(E), B, 0, stream>>>(colp, degf, E);
  k_rsqrt1p<<<g(N), B, 0, stream>>>(degf, dinv, N);
  k_agg1<<<g((long long)E * 32), B, 0, stream>>>(rowp, colp, dinv, bufA, agg, E);
  k_self_relu<<<g((long long)N * FDIM), B, 0, stream>>>(agg, bufA, dinv, b1, N); // agg -> x1

  // 4) scores, gate, sortable keys (x1g overwrites H1 in bufA)
  k_score_gate<<<g((long long)N * 32) , B, 0, stream>>>(agg, wsc, bufA, ukey, N);

  // 5) top-K radix select + index-ordered labels
  for (int p = 0; p < 4; ++p) {
    k_hist<<<g(N), B, 0, stream>>>(ukey, hist, st, p, N);
    k_pick<<<1, 64, 0, stream>>>(st, hist, p);
  }
  k_label<<<1, B, 0, stream>>>(ukey, st, keep, cof, N);

  // 6) cluster assignment (composite-key segment_max equivalents)
  k_deg_row<<<g(E), B, 0, stream>>>(rowp, degr, E);
  k_bestglob1<<<g(N), B, 0, stream>>>(keep, degr, st, N);
  k_bestglob2<<<g(N), B, 0, stream>>>(keep, degr, ukey, st, N);
  k_edge_best<<<g(E), B, 0, stream>>>(rowp, colp, degr, keep, best, E);
  k_fallback<<<1, 64, 0, stream>>>(st, cof);
  k_assign<<<g(N), B, 0, stream>>>(keep, cof, best, st, cid, cnts, N);

  // 7) mean pool
  k_pool<<<g((long long)N * 32), B, 0, stream>>>(cid, bufA, sums, N);
  k_meandiv<<<g((long long)K * FDIM), B, 0, stream>>>(sums, cnts, x_p, K);

  // 8) conv2 on coarse graph (WMMA f32) with edge weights (cu != cv)
  k_gemm128<<<(K + 15) / 16, B, 0, stream>>>(x_p, W2, h2, K);
  k_deg2<<<g(E), B, 0, stream>>>(rowp, colp, cid, deg2, E);
  k_rsqrt1p<<<g(K), B, 0, stream>>>(deg2, dinv2, K);
  k_agg2<<<g((long long)E * 32), B, 0, stream>>>(rowp, colp, cid, dinv2, h2, agg2, E);
  k_self2<<<g((long long)K * FDIM), B, 0, stream>>>(agg2, h2, dinv2, b2, K); // agg2 -> xp2

  // 9) skip GEMM directly into d_out (WMMA f32), then broadcast-add + bias + scalar
  k_gemm128<<<(N + 15) / 16, B, 0, stream>>>(agg, Wskip, out, N);
  k_final<<<g((long long)N * FDIM), B, 0, stream>>>(out, agg2, cid, bskip, N);
}